// surfacePropLoss_15814069584494
// MI455X (gfx1250) — compile-verified
//
#include <hip/hip_runtime.h>
#include <hip/hip_bf16.h>
#include <math.h>

typedef __attribute__((ext_vector_type(2))) float v2f;
typedef __attribute__((ext_vector_type(8))) float v8f;

#define PI_F 3.14159265358979323846f

// Smallest-eigenvalue unit eigenvector of symmetric 3x3 [[xx,xy,xz],[xy,yy,yz],[xz,yz,zz]]
__device__ __forceinline__ void eig3_smallest(
    float xx, float xy, float xz, float yy, float yz, float zz,
    float& nx, float& ny, float& nz)
{
    float q  = (xx + yy + zz) * (1.0f / 3.0f);
    float p1 = xy * xy + xz * xz + yz * yz;
    float a  = xx - q, b = yy - q, c = zz - q;
    float p2 = a * a + b * b + c * c + 2.0f * p1;
    float p  = sqrtf(p2 * (1.0f / 6.0f));
    if (p < 1e-12f) { nx = 1.0f; ny = 0.0f; nz = 0.0f; return; }
    float ip  = 1.0f / p;
    float bxx = a * ip, byy = b * ip, bzz = c * ip;
    float bxy = xy * ip, bxz = xz * ip, byz = yz * ip;
    float detB = bxx * (byy * bzz - byz * byz)
               - bxy * (bxy * bzz - byz * bxz)
               + bxz * (bxy * byz - byy * bxz);
    float r = 0.5f * detB;
    r = fminf(1.0f, fmaxf(-1.0f, r));
    float phi = acosf(r) * (1.0f / 3.0f);
    float lam = q + 2.0f * p * cosf(phi + 2.0f * PI_F / 3.0f); // smallest eigenvalue
    // rows of (A - lam I): r0=(m00,xy,xz) r1=(xy,m11,yz) r2=(xz,yz,m22)
    float m00 = xx - lam, m11 = yy - lam, m22 = zz - lam;
    float c0x = xy * yz - m11 * xz, c0y = xz * xy - m00 * yz, c0z = m00 * m11 - xy * xy; // r0 x r1
    float c1x = xy * m22 - yz * xz, c1y = xz * xz - m00 * m22, c1z = m00 * yz - xy * xz; // r0 x r2
    float c2x = m11 * m22 - yz * yz, c2y = yz * xz - xy * m22, c2z = xy * yz - m11 * xz; // r1 x r2
    float n0 = c0x * c0x + c0y * c0y + c0z * c0z;
    float n1 = c1x * c1x + c1y * c1y + c1z * c1z;
    float n2 = c2x * c2x + c2y * c2y + c2z * c2z;
    float vx = c0x, vy = c0y, vz = c0z, nn = n0;
    if (n1 > nn) { vx = c1x; vy = c1y; vz = c1z; nn = n1; }
    if (n2 > nn) { vx = c2x; vy = c2y; vz = c2z; nn = n2; }
    if (nn < 1e-30f) { nx = 1.0f; ny = 0.0f; nz = 0.0f; return; }
    float inv = rsqrtf(nn);
    nx = vx * inv; ny = vy * inv; nz = vz * inv;
}

// One wave per 32 query rows of one "cloud" (a batch for the global pass, a
// patch for the local pass). Pairwise dot products via V_WMMA_F32_16X16X4_F32
// (3D points zero-padded to K=4). Ranking key r(i,j) = |c_j|^2 - 2*q_i.c_j is
// order-equivalent to squared distance per row.
template <int K>
__global__ __launch_bounds__(32)
void knn_normals_kernel(const float* __restrict__ pts,
                        float* __restrict__ normals,
                        int S, int tiles_per_cloud)
{
    const int lane  = threadIdx.x;       // 0..31
    const int cloud = blockIdx.x / tiles_per_cloud;
    const int tile  = blockIdx.x % tiles_per_cloud;
    const float* cp = pts + (size_t)cloud * (size_t)S * 3;

    __shared__ __align__(16) float s_cand[16][4];  // candidate tile coords (padded row)
    __shared__ __align__(16) float s_cn[16];       // candidate squared norms
    __shared__ __align__(16) float s_dot[32][16];  // dot tile, lane-row layout

    const int m    = lane & 15;
    const int half = lane >> 4;

    // A-tiles (16x4 f32 layout: lanes 0-15 hold K=0,1; lanes 16-31 hold K=2,3)
    int r0 = tile * 32 + m;      if (r0 >= S) r0 = S - 1;
    int r1 = tile * 32 + 16 + m; if (r1 >= S) r1 = S - 1;
    float q0x = cp[3 * r0 + 0], q0y = cp[3 * r0 + 1], q0z = cp[3 * r0 + 2];
    float q1x = cp[3 * r1 + 0], q1y = cp[3 * r1 + 1], q1z = cp[3 * r1 + 2];
    v2f a0, a1;
    a0.x = half ? q0z : q0x; a0.y = half ? 0.0f : q0y;
    a1.x = half ? q1z : q1x; a1.y = half ? 0.0f : q1y;

    // query point owned by this lane (row = tile*32 + lane)
    const int myrow = tile * 32 + lane;
    const int myc   = (myrow < S) ? myrow : S - 1;
    const float qx = cp[3 * myc + 0], qy = cp[3 * myc + 1], qz = cp[3 * myc + 2];

    float dk[K];
#pragma unroll
    for (int i = 0; i < K; ++i) dk[i] = 3.0e38f;

    const int ntiles = (S + 15) >> 4;
    const v8f zero8 = {};
    const float4* rowp = reinterpret_cast<const float4*>(&s_dot[lane][0]);
    const float4* cnp  = reinterpret_cast<const float4*>(&s_cn[0]);

    // ---------------- pass 1: K-th smallest relative distance ----------------
    for (int t = 0; t < ntiles; ++t) {
        const int c0 = t << 4;
        if (lane < 16) {
            int cj = c0 + lane;
            float cx, cy, cz;
            if (cj < S) { cx = cp[3 * cj + 0]; cy = cp[3 * cj + 1]; cz = cp[3 * cj + 2]; }
            else        { cx = 1.0e8f; cy = 1.0e8f; cz = 1.0e8f; }   // pad -> huge distance
            s_cand[lane][0] = cx; s_cand[lane][1] = cy; s_cand[lane][2] = cz;
            s_cn[lane] = cx * cx + cy * cy + cz * cz;
        }
        __builtin_prefetch(cp + 3 * (c0 + 16), 0, 3);
        __syncthreads();
        // B-tile (4x16 f32): load components unconditionally, select per half
        float cb0 = s_cand[m][0], cb1 = s_cand[m][1], cb2 = s_cand[m][2];
        v2f bm;
        bm.x = half ? cb2 : cb0;
        bm.y = half ? 0.0f : cb1;
        v8f d0 = __builtin_amdgcn_wmma_f32_16x16x4_f32(false, a0, false, bm, (short)0, zero8, false, false);
        v8f d1 = __builtin_amdgcn_wmma_f32_16x16x4_f32(false, a1, false, bm, (short)0, zero8, false, false);
#pragma unroll
        for (int v = 0; v < 8; ++v) {
            s_dot[v + 8 * half][m]      = d0[v];
            s_dot[16 + v + 8 * half][m] = d1[v];
        }
        __syncthreads();
#pragma unroll 1
        for (int j4 = 0; j4 < 4; ++j4) {
            float4 dd = rowp[j4];
            float4 cc = cnp[j4];
            float rr[4];
            rr[0] = fmaf(-2.0f, dd.x, cc.x);
            rr[1] = fmaf(-2.0f, dd.y, cc.y);
            rr[2] = fmaf(-2.0f, dd.z, cc.z);
            rr[3] = fmaf(-2.0f, dd.w, cc.w);
#pragma unroll
            for (int u = 0; u < 4; ++u) {
                float r = rr[u];
                if (r < dk[K - 1]) {           // sorted insert (unrolled shift)
#pragma unroll
                    for (int s = K - 1; s > 0; --s) {
                        float shifted = (r < dk[s - 1]) ? dk[s - 1] : r;
                        if (r < dk[s]) dk[s] = shifted;
                    }
                    if (r < dk[0]) dk[0] = r;
                }
            }
        }
        __syncthreads();
    }

    const float thr = dk[K - 1];

    // -------- pass 2: covariance of centered kNN (bitwise-identical r) -------
    float cxx = 0.f, cxy = 0.f, cxz = 0.f, cyy = 0.f, cyz = 0.f, czz = 0.f;
    for (int t = 0; t < ntiles; ++t) {
        const int c0 = t << 4;
        if (lane < 16) {
            int cj = c0 + lane;
            float cx, cy, cz;
            if (cj < S) { cx = cp[3 * cj + 0]; cy = cp[3 * cj + 1]; cz = cp[3 * cj + 2]; }
            else        { cx = 1.0e8f; cy = 1.0e8f; cz = 1.0e8f; }
            s_cand[lane][0] = cx; s_cand[lane][1] = cy; s_cand[lane][2] = cz;
            s_cn[lane] = cx * cx + cy * cy + cz * cz;
        }
        __builtin_prefetch(cp + 3 * (c0 + 16), 0, 3);
        __syncthreads();
        float cb0 = s_cand[m][0], cb1 = s_cand[m][1], cb2 = s_cand[m][2];
        v2f bm;
        bm.x = half ? cb2 : cb0;
        bm.y = half ? 0.0f : cb1;
        v8f d0 = __builtin_amdgcn_wmma_f32_16x16x4_f32(false, a0, false, bm, (short)0, zero8, false, false);
        v8f d1 = __builtin_amdgcn_wmma_f32_16x16x4_f32(false, a1, false, bm, (short)0, zero8, false, false);
#pragma unroll
        for (int v = 0; v < 8; ++v) {
            s_dot[v + 8 * half][m]      = d0[v];
            s_dot[16 + v + 8 * half][m] = d1[v];
        }
        __syncthreads();
#pragma unroll 1
        for (int j4 = 0; j4 < 4; ++j4) {
            float4 dd = rowp[j4];
            float4 cc = cnp[j4];
            float rr[4];
            rr[0] = fmaf(-2.0f, dd.x, cc.x);
            rr[1] = fmaf(-2.0f, dd.y, cc.y);
            rr[2] = fmaf(-2.0f, dd.z, cc.z);
            rr[3] = fmaf(-2.0f, dd.w, cc.w);
#pragma unroll
            for (int u = 0; u < 4; ++u) {
                if (rr[u] <= thr) {
                    int j = j4 * 4 + u;
                    float dx = s_cand[j][0] - qx;
                    float dy = s_cand[j][1] - qy;
                    float dz = s_cand[j][2] - qz;
                    cxx = fmaf(dx, dx, cxx); cxy = fmaf(dx, dy, cxy); cxz = fmaf(dx, dz, cxz);
                    cyy = fmaf(dy, dy, cyy); cyz = fmaf(dy, dz, cyz); czz = fmaf(dz, dz, czz);
                }
            }
        }
        __syncthreads();
    }

    float nx, ny, nz;
    eig3_smallest(cxx, cxy, cxz, cyy, cyz, czz, nx, ny, nz);
    if (myrow < S) {
        float* o = normals + (size_t)(cloud * S + myrow) * 3;
        o[0] = nx; o[1] = ny; o[2] = nz;
    }
}

__global__ __launch_bounds__(256)
void loss_partial_kernel(const float* __restrict__ ng,
                         const float* __restrict__ np,
                         float* __restrict__ partials, int npts)
{
    __shared__ float red[256];
    int i = blockIdx.x * 256 + threadIdx.x;
    float v = 0.0f;
    if (i < npts) {
        float d0 = fabsf(np[3 * i + 0]) - fabsf(ng[3 * i + 0]);
        float d1 = fabsf(np[3 * i + 1]) - fabsf(ng[3 * i + 1]);
        float d2 = fabsf(np[3 * i + 2]) - fabsf(ng[3 * i + 2]);
        v = sqrtf(d0 * d0 + d1 * d1 + d2 * d2);
    }
    red[threadIdx.x] = v;
    __syncthreads();
#pragma unroll
    for (int s = 128; s > 0; s >>= 1) {
        if (threadIdx.x < s) red[threadIdx.x] += red[threadIdx.x + s];
        __syncthreads();
    }
    if (threadIdx.x == 0) partials[blockIdx.x] = red[0];
}

__global__ __launch_bounds__(256)
void loss_final_kernel(const float* __restrict__ partials, int n,
                       float* __restrict__ out, float inv_n)
{
    __shared__ float red[256];
    float v = 0.0f;
    for (int i = threadIdx.x; i < n; i += 256) v += partials[i];
    red[threadIdx.x] = v;
    __syncthreads();
#pragma unroll
    for (int s = 128; s > 0; s >>= 1) {
        if (threadIdx.x < s) red[threadIdx.x] += red[threadIdx.x + s];
        __syncthreads();
    }
    if (threadIdx.x == 0) out[0] = red[0] * inv_n;
}

extern "C" void kernel_launch(void* const* d_in, const int* in_sizes, int n_in,
                              void* d_out, int out_size, void* d_ws, size_t ws_size,
                              hipStream_t stream)
{
    (void)n_in; (void)out_size; (void)ws_size;
    const float* pc = (const float*)d_in[0];   // [B, N, 3] float32 ; d_in[1] unused

    const int B = 8;
    const int npts = in_sizes[0] / 3;          // B*N = 20000
    const int N = npts / B;                    // 2500
    const int NP = 25;                         // patches per cloud
    const int P = N / NP;                      // 100 points per patch

    float* ws       = (float*)d_ws;
    float* ng       = ws;                      // npts*3 floats
    float* npatch   = ws + (size_t)npts * 3;   // npts*3 floats
    float* partials = ws + (size_t)npts * 6;   // reduction partials

    // Global normals: B clouds of N points, k = 32
    {
        int tiles = (N + 31) / 32;
        knn_normals_kernel<32><<<dim3(B * tiles), dim3(32), 0, stream>>>(pc, ng, N, tiles);
    }
    // Patch normals: B*NP clouds of P points, k = 16 (flat layout matches reshape)
    {
        int tiles = (P + 31) / 32;
        knn_normals_kernel<16><<<dim3(B * NP * tiles), dim3(32), 0, stream>>>(pc, npatch, P, tiles);
    }
    // Mean of per-point ||abs(n_patch) - abs(n_global)||_2
    int nb = (npts + 255) / 256;
    loss_partial_kernel<<<dim3(nb), dim3(256), 0, stream>>>(ng, npatch, partials, npts);
    loss_final_kernel<<<dim3(1), dim3(256), 0, stream>>>(partials, nb, (float*)d_out, 1.0f / (float)npts);
}